// SoftsplatLite_283467842030
// MI455X (gfx1250) — compile-verified
//
#include <hip/hip_runtime.h>

// Softmax splatting (softsplat, mode='soft') for MI455X / gfx1250.
// B=8, C=3, H=W=1024 fixed by the reference.
//
// Scatter with native non-returning f32 atomics (verified lowering:
// vmem count matches global_atomic_add_f32, no CAS loop). Accumulation state
// (d_out 96 MiB + 32 MiB norm plane) fits in the 192 MB L2, so atomics
// resolve at L2 rate; HBM traffic ~420 MB -> ~18 us at 23.3 TB/s.
// WMMA does not apply: forward splatting has no GEMM structure.

constexpr int B = 8;
constexpr int C = 3;
constexpr int H = 1024;
constexpr int W = 1024;
constexpr int HW = H * W;

// Non-returning agent-scope float atomic add -> global_atomic_add_f32
// (fire-and-forget, tracked by STOREcnt).
__device__ __forceinline__ void gadd(float* p, float v) {
    (void)__hip_atomic_fetch_add(p, v, __ATOMIC_RELAXED, __HIP_MEMORY_SCOPE_AGENT);
}

__global__ __launch_bounds__(256) void splat_kernel(
    const float* __restrict__ img,     // [B,3,H,W]
    const float* __restrict__ flow,    // [B,2,H,W]
    const float* __restrict__ metric,  // [B,1,H,W]
    float* __restrict__ acc,           // [B,3,H,W]  (= d_out, pre-zeroed)
    float* __restrict__ nrm)           // [B,H,W]    (in d_ws, pre-zeroed)
{
    const int idx = blockIdx.x * blockDim.x + threadIdx.x;  // over B*HW
    if (idx >= B * HW) return;
    const int b = idx / HW;
    const int p = idx - b * HW;
    const int y = p / W;
    const int x = p - y * W;

    // Gather inputs (plane-coalesced b32 loads; compiler forms clauses).
    const float fxv = flow[(b * 2 + 0) * HW + p] + (float)x;
    const float fyv = flow[(b * 2 + 1) * HW + p] + (float)y;
    const float wgt = expf(metric[b * HW + p]);
    const float vr = img[(b * 3 + 0) * HW + p] * wgt;
    const float vg = img[(b * 3 + 1) * HW + p] * wgt;
    const float vb = img[(b * 3 + 2) * HW + p] * wgt;

    const float x0f = floorf(fxv);
    const float y0f = floorf(fyv);
    const int   x0  = (int)x0f;
    const int   y0  = (int)y0f;
    const float ax  = fxv - x0f;   // in [0,1)
    const float ay  = fyv - y0f;

    const float wx[2] = {1.0f - ax, ax};
    const float wy[2] = {1.0f - ay, ay};

    float* __restrict__ accB = acc + (size_t)b * 3 * HW;
    float* __restrict__ nrmB = nrm + (size_t)b * HW;

    if (x0 >= 0 && y0 >= 0 && x0 + 1 < W && y0 + 1 < H) {
        // Interior fast path (~98% of pixels): 16 unconditional atomics.
        const int q00 = y0 * W + x0;
#pragma unroll
        for (int dy = 0; dy < 2; ++dy) {
#pragma unroll
            for (int dx = 0; dx < 2; ++dx) {
                const float w = wx[dx] * wy[dy];
                const int   q = q00 + dy * W + dx;
                gadd(accB + 0 * HW + q, vr * w);
                gadd(accB + 1 * HW + q, vg * w);
                gadd(accB + 2 * HW + q, vb * w);
                gadd(nrmB + q,          wgt * w);
            }
        }
    } else {
        // Border path: skip out-of-bounds taps (reference adds w=0 there: no-op).
#pragma unroll
        for (int dy = 0; dy < 2; ++dy) {
            const int yi = y0 + dy;
            if (yi < 0 || yi >= H) continue;
#pragma unroll
            for (int dx = 0; dx < 2; ++dx) {
                const int xi = x0 + dx;
                if (xi < 0 || xi >= W) continue;
                const float w = wx[dx] * wy[dy];
                const int   q = yi * W + xi;
                gadd(accB + 0 * HW + q, vr * w);
                gadd(accB + 1 * HW + q, vg * w);
                gadd(accB + 2 * HW + q, vb * w);
                gadd(nrmB + q,          wgt * w);
            }
        }
    }
}

// Normalize 4 pixels per thread with b128 traffic. W % 4 == 0, planes are
// 16B-aligned, so float4 access is safe.
__global__ __launch_bounds__(256) void normalize_kernel(
    float* __restrict__ out,          // [B,3,H,W] accumulated, normalized in place
    const float* __restrict__ nrm)    // [B,H,W]
{
    constexpr int HW4 = HW / 4;
    const int idx = blockIdx.x * blockDim.x + threadIdx.x;  // over B*HW4
    if (idx >= B * HW4) return;
    const int b  = idx / HW4;
    const int p4 = idx - b * HW4;

    const float4* __restrict__ nrm4 = (const float4*)nrm;
    float4 n = nrm4[(size_t)b * HW4 + p4];
    n.x = (n.x == 0.0f) ? 1.0f : n.x;
    n.y = (n.y == 0.0f) ? 1.0f : n.y;
    n.z = (n.z == 0.0f) ? 1.0f : n.z;
    n.w = (n.w == 0.0f) ? 1.0f : n.w;

    float4* __restrict__ out4 = (float4*)out;
#pragma unroll
    for (int c = 0; c < C; ++c) {
        const size_t o = ((size_t)b * 3 + c) * HW4 + p4;
        float4 v = out4[o];
        // IEEE divide per element, matching the reference's acc / norm.
        v.x = v.x / n.x;
        v.y = v.y / n.y;
        v.z = v.z / n.z;
        v.w = v.w / n.w;
        out4[o] = v;
    }
}

extern "C" void kernel_launch(void* const* d_in, const int* in_sizes, int n_in,
                              void* d_out, int out_size, void* d_ws, size_t ws_size,
                              hipStream_t stream) {
    const float* img    = (const float*)d_in[0];
    const float* flow   = (const float*)d_in[1];
    const float* metric = (const float*)d_in[2];
    float* acc = (float*)d_out;         // [B,3,H,W] accumulator == final output
    float* nrm = (float*)d_ws;          // [B,H,W] weight accumulator (32 MiB)

    // Zero accumulators every call (harness poisons buffers; must be
    // deterministic). hipMemsetAsync is graph-capture safe.
    hipMemsetAsync(acc, 0, (size_t)B * 3 * HW * sizeof(float), stream);
    hipMemsetAsync(nrm, 0, (size_t)B * HW * sizeof(float), stream);

    const int threads = 256;                      // 8 waves on wave32
    const int nSplat = B * HW;
    const int blocksSplat = (nSplat + threads - 1) / threads;
    splat_kernel<<<blocksSplat, threads, 0, stream>>>(img, flow, metric, acc, nrm);

    const int nNorm = B * (HW / 4);
    const int blocksNorm = (nNorm + threads - 1) / threads;
    normalize_kernel<<<blocksNorm, threads, 0, stream>>>(acc, nrm);
}